// HyperTurbulentGenerator_75866302316778
// MI455X (gfx1250) — compile-verified
//
#include <hip/hip_runtime.h>

typedef __attribute__((ext_vector_type(2))) float v2f;
typedef __attribute__((ext_vector_type(8))) float v8f;

#define B_  64
#define S_  512
#define D_  32
#define NB_ 8

// ---------------------------------------------------------------------------
// Kernel 1: A[b] = (K_sum - K_sum^T) - R_sum^T @ R_sum   (64 tiny matrices)
// ---------------------------------------------------------------------------
__global__ __launch_bounds__(256) void ht_buildA(
    const float* __restrict__ k_coeffs, const float* __restrict__ r_coeffs,
    const float* __restrict__ K_bases,  const float* __restrict__ R_bases,
    float* __restrict__ A_out)
{
    __shared__ float Ks[D_ * D_];
    __shared__ float Rs[D_ * D_];
    const int b   = blockIdx.x;
    const int tid = threadIdx.x;

    float kc[NB_], rc[NB_];
#pragma unroll
    for (int n = 0; n < NB_; ++n) {
        kc[n] = k_coeffs[b * NB_ + n];
        rc[n] = r_coeffs[b * NB_ + n];
    }
#pragma unroll
    for (int e = tid; e < D_ * D_; e += 256) {
        float ks = 0.f, rs = 0.f;
#pragma unroll
        for (int n = 0; n < NB_; ++n) {
            ks += kc[n] * K_bases[n * D_ * D_ + e];
            rs += rc[n] * R_bases[n * D_ * D_ + e];
        }
        Ks[e] = ks; Rs[e] = rs;
    }
    __syncthreads();
#pragma unroll
    for (int e = tid; e < D_ * D_; e += 256) {
        const int i = e >> 5, j = e & 31;
        float g = 0.f;
#pragma unroll
        for (int k = 0; k < D_; ++k) g += Rs[k * D_ + i] * Rs[k * D_ + j];
        A_out[b * D_ * D_ + e] = Ks[i * D_ + j] - Ks[j * D_ + i] - g;
    }
}

// ---------------------------------------------------------------------------
// 32x32x32 fp32 matmul helpers built on V_WMMA_F32_16X16X4_F32 (wave32).
// All matrices are LDS, row-major, ld = 32. Block = 4 waves; wave w owns
// output tile (mi, ni) = (w>>1, w&1); 8 K-steps of K=4 each.
// A frag (ISA 32-bit A 16x4): lanes 0-15 -> rows, v0/v1 = K 0/1 of group;
//                             lanes 16-31 -> same rows, K 2/3.
// B/C/D frags: rows striped across lanes within a VGPR (ISA 7.12.2).
// ---------------------------------------------------------------------------

// Load this wave's 8 A-fragments (one per K-step) into registers.
__device__ inline void loadA_frags(const float* __restrict__ A,
                                   v2f (&af)[8], int mi, int half, int l15)
{
    const int arow = mi * 16 + l15;
#pragma unroll
    for (int kk = 0; kk < 8; ++kk) {
        const int k0 = kk * 4 + 2 * half;
        af[kk] = *(const v2f*)(A + arow * 32 + k0);   // 8B-aligned pair
    }
}

// dst = scale * (Afrags @ Bm) (+ I if addI); A operand from registers.
__device__ inline void mm32_wmma_regA(float* __restrict__ dst,
                                      const v2f (&af)[8],
                                      const float* __restrict__ Bm,
                                      float scale, int addI,
                                      int mi, int ni, int half, int l15)
{
    const int bcol = ni * 16 + l15;
    v8f c = {};
#pragma unroll
    for (int kk = 0; kk < 8; ++kk) {
        const int k0 = kk * 4 + 2 * half;
        v2f bb;
        bb.x = Bm[(k0    ) * 32 + bcol];
        bb.y = Bm[(k0 + 1) * 32 + bcol];
        c = __builtin_amdgcn_wmma_f32_16x16x4_f32(
                false, af[kk], false, bb, (short)0, c, false, false);
    }
#pragma unroll
    for (int v = 0; v < 8; ++v) {
        const int row = mi * 16 + half * 8 + v;
        float val = c[v] * scale;
        if (addI && row == bcol) val += 1.0f;
        dst[row * 32 + bcol] = val;
    }
}

// dst = A @ Bm; A operand loaded from LDS (used when A changes per call).
__device__ inline void mm32_wmma_ldsA(float* __restrict__ dst,
                                      const float* __restrict__ A,
                                      const float* __restrict__ Bm,
                                      int mi, int ni, int half, int l15)
{
    const int arow = mi * 16 + l15;
    const int bcol = ni * 16 + l15;
    v8f c = {};
#pragma unroll
    for (int kk = 0; kk < 8; ++kk) {
        const int k0 = kk * 4 + 2 * half;
        v2f a = *(const v2f*)(A + arow * 32 + k0);
        v2f bb;
        bb.x = Bm[(k0    ) * 32 + bcol];
        bb.y = Bm[(k0 + 1) * 32 + bcol];
        c = __builtin_amdgcn_wmma_f32_16x16x4_f32(
                false, a, false, bb, (short)0, c, false, false);
    }
#pragma unroll
    for (int v = 0; v < 8; ++v) {
        const int row = mi * 16 + half * 8 + v;
        dst[row * 32 + bcol] = c[v];
    }
}

// ---------------------------------------------------------------------------
// Kernel 2: per (b,s): X = t*A[b]; scaling-squaring Taylor expm; out = P @ z0
// grid (S, B), 128 threads (4 waves).
// ---------------------------------------------------------------------------
__global__ __launch_bounds__(128) void ht_expm_apply(
    const float* __restrict__ A_all, const float* __restrict__ time_steps,
    const float* __restrict__ z0,    float* __restrict__ out)
{
    __shared__ float X [D_ * D_];
    __shared__ float T0[D_ * D_];
    __shared__ float T1[D_ * D_];
    __shared__ float zsh[D_];
    __shared__ float rowsum[D_];
    __shared__ int   jshared;
    __shared__ float sshared;

    const int s    = blockIdx.x;
    const int b    = blockIdx.y;
    const int tid  = threadIdx.x;
    const int wave = tid >> 5;
    const int lane = tid & 31;
    const int mi   = (wave >> 1) & 1;
    const int ni   = wave & 1;
    const int half = lane >> 4;
    const int l15  = lane & 15;

    const float  t  = time_steps[s];
    const float* Ag = A_all + b * D_ * D_;

#pragma unroll
    for (int e = tid; e < D_ * D_; e += 128) X[e] = t * Ag[e];
    if (tid < D_) zsh[tid] = z0[b * D_ + tid];
    __syncthreads();

    // inf-norm -> scaling power j so that ||X/2^j||_inf <= 0.25
    if (tid < D_) {
        float rs = 0.f;
#pragma unroll
        for (int k = 0; k < D_; ++k) rs += fabsf(X[tid * D_ + k]);
        rowsum[tid] = rs;
    }
    __syncthreads();
    if (tid == 0) {
        float nm = 0.f;
#pragma unroll
        for (int k = 0; k < D_; ++k) nm = fmaxf(nm, rowsum[k]);
        int j = 0;
        if (nm > 0.25f) {
            j = (int)ceilf(log2f(nm * 4.0f));
            j = j < 0 ? 0 : (j > 30 ? 30 : j);
        }
        jshared = j;
        sshared = exp2f((float)(-j));
    }
    __syncthreads();
    const int   jsq = jshared;
    const float sc  = sshared;

#pragma unroll
    for (int e = tid; e < D_ * D_; e += 128) X[e] *= sc;
    // Taylor order 8, Horner seed: T = I + X/8
#pragma unroll
    for (int e = tid; e < D_ * D_; e += 128) {
        const int r = e >> 5, cidx = e & 31;
        T0[e] = X[e] * 0.125f + ((r == cidx) ? 1.0f : 0.0f);
    }
    __syncthreads();

    // X is invariant across the whole Horner chain: keep this wave's A
    // fragments in 16 VGPRs and halve LDS traffic for 7 of ~10 matmuls.
    v2f afragX[8];
    loadA_frags(X, afragX, mi, half, l15);

    float* Tsrc = T0;
    float* Tdst = T1;
#pragma unroll
    for (int k = 7; k >= 1; --k) {        // T = I + (X @ T) / k
        mm32_wmma_regA(Tdst, afragX, Tsrc, 1.0f / (float)k, 1,
                       mi, ni, half, l15);
        __syncthreads();
        float* tmp = Tsrc; Tsrc = Tdst; Tdst = tmp;
    }
    for (int it = 0; it < jsq; ++it) {    // P = P @ P, j times
        mm32_wmma_ldsA(Tdst, Tsrc, Tsrc, mi, ni, half, l15);
        __syncthreads();
        float* tmp = Tsrc; Tsrc = Tdst; Tdst = tmp;
    }

    // out[b,s,o] = sum_i P[o,i] * z0[b,i]
    if (tid < D_) {
        float acc = 0.f;
#pragma unroll
        for (int i = 0; i < D_; ++i) acc += Tsrc[tid * D_ + i] * zsh[i];
        out[(b * S_ + s) * D_ + tid] = acc;
    }
}

// ---------------------------------------------------------------------------
extern "C" void kernel_launch(void* const* d_in, const int* in_sizes, int n_in,
                              void* d_out, int out_size, void* d_ws, size_t ws_size,
                              hipStream_t stream)
{
    const float* z0 = (const float*)d_in[0];
    const float* ts = (const float*)d_in[1];
    const float* kc = (const float*)d_in[2];
    const float* rc = (const float*)d_in[3];
    const float* Kb = (const float*)d_in[4];
    const float* Rb = (const float*)d_in[5];
    float* out = (float*)d_out;
    float* Aws = (float*)d_ws;   // 64 * 1024 floats = 256 KB

    ht_buildA<<<B_, 256, 0, stream>>>(kc, rc, Kb, Rb, Aws);
    dim3 grid(S_, B_);
    ht_expm_apply<<<grid, 128, 0, stream>>>(Aws, ts, z0, out);
}